// PairwiseInteraction_80882824118358
// MI455X (gfx1250) — compile-verified
//
#include <hip/hip_runtime.h>
#include <hip/hip_bf16.h>
#include <math.h>

#define NB   128   // batches
#define NK   32    // slots per batch
#define ND   256   // feature dim
#define NH   128   // hidden dim

typedef __attribute__((ext_vector_type(2))) float v2f;
typedef __attribute__((ext_vector_type(4))) float v4f;
typedef __attribute__((ext_vector_type(8))) float v8f;

// padded LDS leading dims (pad=4 floats -> 16-row strided f32-WMMA A-loads are bank-conflict-free)
#define LDS_S 260   // slots panel: 32 x 256
#define LDP   132   // 32 x 128 panels

__device__ __forceinline__ v8f wmma_f32(v2f a, v2f b, v8f c) {
  // V_WMMA_F32_16X16X4_F32 : D = A(16x4) x B(4x16) + C(16x16), all fp32
  return __builtin_amdgcn_wmma_f32_16x16x4_f32(false, a, false, b, (short)0, c, false, false);
}

__global__ __launch_bounds__(128)
void pairwise_fused(const float* __restrict__ slots,
                    const float* __restrict__ W1, const float* __restrict__ b1,
                    const float* __restrict__ W2, const float* __restrict__ b2,
                    const float* __restrict__ Wq, const float* __restrict__ bq,
                    const float* __restrict__ Wk, const float* __restrict__ bk,
                    float* __restrict__ out) {
  extern __shared__ float smem[];
  float* sS  = smem;                 // 32*260 : slots[b]   (reused later as g)
  float* sPA = smem + 32 * LDS_S;    // 32*132 : q, later pi(+b1)
  float* sPB = sPA + 32 * LDP;       // 32*132 : k, later pj
  float* sL  = sPB + 32 * LDP;       // 32*32  : logits -> attn
  float* sG  = sS;                   // 32*132 : attn-weighted relu sums (overlaps sS)

  const int tid  = threadIdx.x;
  const int wid  = tid >> 5;         // 4 waves
  const int lane = tid & 31;
  const int r16  = lane & 15;        // 16-dim index within tile
  const int kh   = lane >> 4;        // K-half select for f32 WMMA operands
  const int b    = blockIdx.x;

  // ---------- Phase 1: stage slots[b] (32x256 f32) in LDS ----------
  const float* sb = slots + (size_t)b * NK * ND;
  for (int e = tid; e < (NK * ND) / 4; e += blockDim.x) {
    int flat = e * 4;
    int row  = flat >> 8;       // /256
    int col  = flat & 255;
    *(v4f*)(sS + row * LDS_S + col) = *(const v4f*)(sb + flat);
  }
  __syncthreads();

  // ---------- Phase 2a: q = slots@Wq + bq ; k = slots@Wk + bk ----------
  // 16 N-tile combos (8 for q, 8 for k); each wave does 4, with 2 M-tile accumulators sharing B.
  for (int cmb = wid; cmb < 16; cmb += 4) {
    const int isK  = cmb >> 3;
    const int col0 = (cmb & 7) * 16;
    const float* Wm = isK ? Wk : Wq;          // (256 x 128) row-major
    v8f c0 = {}; v8f c1 = {};
    for (int kk = 0; kk < ND; kk += 4) {
      v2f bv;
      bv.x = Wm[(kk + 2 * kh    ) * NH + col0 + r16];
      bv.y = Wm[(kk + 2 * kh + 1) * NH + col0 + r16];
      v2f a0 = *(const v2f*)(sS + r16 * LDS_S + kk + 2 * kh);
      v2f a1 = *(const v2f*)(sS + (16 + r16) * LDS_S + kk + 2 * kh);
      c0 = wmma_f32(a0, bv, c0);
      c1 = wmma_f32(a1, bv, c1);
    }
    const float bias = isK ? bk[col0 + r16] : bq[col0 + r16];
    float* dst = isK ? sPB : sPA;
    #pragma unroll
    for (int v = 0; v < 8; ++v) {
      dst[(v + 8 * kh) * LDP + col0 + r16]      = c0[v] + bias;
      dst[(16 + v + 8 * kh) * LDP + col0 + r16] = c1[v] + bias;
    }
  }
  __syncthreads();

  // ---------- Phase 3a: logits = scale * q k^T, diagonal mask ----------
  {
    const int mt = wid >> 1, nt = wid & 1;    // 4 tiles of 16x16, one per wave
    v8f c = {};
    for (int kk = 0; kk < NH; kk += 4) {
      v2f a  = *(const v2f*)(sPA + (mt * 16 + r16) * LDP + kk + 2 * kh);
      v2f bv = *(const v2f*)(sPB + (nt * 16 + r16) * LDP + kk + 2 * kh); // k row == (k^T) col
      c = wmma_f32(a, bv, c);
    }
    const float scale = 0.08838834764831845f; // 128^-0.5
    #pragma unroll
    for (int v = 0; v < 8; ++v) {
      int row = mt * 16 + v + 8 * kh;
      int col = nt * 16 + r16;
      float x = c[v] * scale;
      if (row == col) x = -1000000000.0f;
      sL[row * NK + col] = x;
    }
  }
  __syncthreads();

  // ---------- Phase 2b: pi = slots@W1i + b1 ; pj = slots@W1j (overwrite q/k panels) ----------
  for (int cmb = wid; cmb < 16; cmb += 4) {
    const int isJ  = cmb >> 3;
    const int col0 = (cmb & 7) * 16;
    const float* Wm = W1 + (size_t)isJ * ND * NH;   // W1 rows [0:256) / [256:512)
    v8f c0 = {}; v8f c1 = {};
    for (int kk = 0; kk < ND; kk += 4) {
      v2f bv;
      bv.x = Wm[(kk + 2 * kh    ) * NH + col0 + r16];
      bv.y = Wm[(kk + 2 * kh + 1) * NH + col0 + r16];
      v2f a0 = *(const v2f*)(sS + r16 * LDS_S + kk + 2 * kh);
      v2f a1 = *(const v2f*)(sS + (16 + r16) * LDS_S + kk + 2 * kh);
      c0 = wmma_f32(a0, bv, c0);
      c1 = wmma_f32(a1, bv, c1);
    }
    const float bias = isJ ? 0.0f : b1[col0 + r16]; // fold b1 into pi
    float* dst = isJ ? sPB : sPA;
    #pragma unroll
    for (int v = 0; v < 8; ++v) {
      dst[(v + 8 * kh) * LDP + col0 + r16]      = c0[v] + bias;
      dst[(16 + v + 8 * kh) * LDP + col0 + r16] = c1[v] + bias;
    }
  }
  __syncthreads();

  // ---------- Phase 3b: row softmax of logits (32 rows) ----------
  if (tid < NK) {
    float mx = -1e30f;
    for (int j = 0; j < NK; ++j) mx = fmaxf(mx, sL[tid * NK + j]);
    float s = 0.0f;
    for (int j = 0; j < NK; ++j) {
      float e = __expf(sL[tid * NK + j] - mx);
      sL[tid * NK + j] = e;
      s += e;
    }
    float inv = 1.0f / s;
    for (int j = 0; j < NK; ++j) sL[tid * NK + j] *= inv;
  }
  __syncthreads();

  // ---------- Phase 4: g[i,h] = sum_j attn[i,j] * relu(pi[i,h] + pj[j,h]) ----------
  // (softmax rows sum to 1, so  out = g@W2 + b2  exactly matches the reference)
  {
    const int h = tid;                       // 128 threads == 128 hidden channels
    for (int i = 0; i < NK; ++i) {
      float pih = sPA[i * LDP + h];
      float acc = 0.0f;
      #pragma unroll 8
      for (int j = 0; j < NK; ++j)
        acc += sL[i * NK + j] * fmaxf(pih + sPB[j * LDP + h], 0.0f);
      sG[i * LDP + h] = acc;
    }
  }
  __syncthreads();

  // ---------- Phase 5: out[b] = g @ W2 + b2  (32x128 @ 128x256) ----------
  float* ob = out + (size_t)b * NK * ND;
  for (int nt = wid; nt < 16; nt += 4) {
    const int col0 = nt * 16;
    v8f c0 = {}; v8f c1 = {};
    for (int kk = 0; kk < NH; kk += 4) {
      v2f bv;
      bv.x = W2[(kk + 2 * kh    ) * ND + col0 + r16];
      bv.y = W2[(kk + 2 * kh + 1) * ND + col0 + r16];
      v2f a0 = *(const v2f*)(sG + r16 * LDP + kk + 2 * kh);
      v2f a1 = *(const v2f*)(sG + (16 + r16) * LDP + kk + 2 * kh);
      c0 = wmma_f32(a0, bv, c0);
      c1 = wmma_f32(a1, bv, c1);
    }
    const float bias = b2[col0 + r16];
    #pragma unroll
    for (int v = 0; v < 8; ++v) {
      ob[(v + 8 * kh) * ND + col0 + r16]      = c0[v] + bias;
      ob[(16 + v + 8 * kh) * ND + col0 + r16] = c1[v] + bias;
    }
  }
}

extern "C" void kernel_launch(void* const* d_in, const int* in_sizes, int n_in,
                              void* d_out, int out_size, void* d_ws, size_t ws_size,
                              hipStream_t stream) {
  const float* slots = (const float*)d_in[0];
  const float* W1    = (const float*)d_in[1];
  const float* b1    = (const float*)d_in[2];
  const float* W2    = (const float*)d_in[3];
  const float* b2    = (const float*)d_in[4];
  const float* Wq    = (const float*)d_in[5];
  const float* bq    = (const float*)d_in[6];
  const float* Wk    = (const float*)d_in[7];
  const float* bk    = (const float*)d_in[8];
  float* outp = (float*)d_out;

  const size_t shmem = (size_t)(32 * LDS_S + 2 * 32 * LDP + NK * NK) * sizeof(float); // ~70 KB
  pairwise_fused<<<NB, 128, shmem, stream>>>(slots, W1, b1, W2, b2, Wq, bq, Wk, bk, outp);
}